// SemanticGuidedGate_5471788335642
// MI455X (gfx1250) — compile-verified
//
#include <hip/hip_runtime.h>
#include <hip/hip_bf16.h>
#include <math.h>

// Problem constants (from reference)
#define BB 4
#define KK 16
#define NSRC 4096
#define NTGT 16384

typedef __attribute__((ext_vector_type(2))) float v2f;
typedef __attribute__((ext_vector_type(4))) float v4f;
typedef __attribute__((ext_vector_type(8))) float v8f;

// ---------------------------------------------------------------------------
// Pass 1: per-source precompute.
//   feat[b,s]     = max_k logits[b,k,s]
//   packed[b,s,:] = { x, y, z, x^2+y^2+z^2 }   (one 16B record per source)
// ---------------------------------------------------------------------------
__global__ __launch_bounds__(256) void sgg_precompute(
    const float* __restrict__ logits,   // [B, K, NS]
    const float* __restrict__ spos,     // [B, 3, NS]
    float* __restrict__ feat,           // [B, NS]
    v4f*   __restrict__ packed)         // [B, NS] 16B records
{
    int i = blockIdx.x * blockDim.x + threadIdx.x;     // b*NS + s
    if (i >= BB * NSRC) return;
    int b = i >> 12;                                   // / NSRC
    int s = i & (NSRC - 1);

    const float* L = logits + ((size_t)b * KK) * NSRC + s;
    float mx = L[0];
#pragma unroll
    for (int k = 1; k < KK; ++k) mx = fmaxf(mx, L[(size_t)k * NSRC]);
    feat[i] = mx;

    const float* S = spos + ((size_t)b * 3) * NSRC + s;
    float x = S[0], y = S[NSRC], z = S[2 * NSRC];
    v4f p;
    p.x = x; p.y = y; p.z = z; p.w = x * x + y * y + z * z;
    packed[i] = p;
}

// ---------------------------------------------------------------------------
// Pass 2: one wave32 per 16-target tile.
// score(M,N) = |s_N|^2 - 2 * t_M . s_N  in ONE V_WMMA_F32_16X16X4_F32 with
// C = inline 0, by folding the |s|^2 bias into the spare K=3 slot:
//   A (16x4): K=0..2 = -2*t coords, K=3 = 1.0
//   B (4x16): K=0..2 =  s coords,   K=3 = |s|^2
// One b128 load per lane per 16-source chunk; running (min, argmin) kept in
// the C/D register layout; shfl_xor butterfly reduce; sigmoid-gather write.
// ---------------------------------------------------------------------------
__global__ __launch_bounds__(256) void sgg_nn_gate(
    const float* __restrict__ tpos,     // [B, 3, NT]
    const v4f*   __restrict__ packed,   // [B, NS] {x,y,z,|s|^2}
    const float* __restrict__ feat,     // [B, NS]
    float* __restrict__ out)            // [B, NT]
{
    const int gwave = (blockIdx.x * blockDim.x + threadIdx.x) >> 5;  // tile id
    const int lane  = threadIdx.x & 31;
    const int b     = gwave >> 10;                 // NT/16 = 1024 tiles/batch
    const int t0    = (gwave & 1023) << 4;         // first target row of tile
    const int m     = lane & 15;                   // row (A) / col (B,C,D)
    const bool hi   = lane >= 16;

    const float* tp = tpos   + (size_t)b * 3 * NTGT;
    const v4f*   pk = packed + (size_t)b * NSRC + m;   // per-lane base
    const float* ft = feat   + (size_t)b * NSRC;

    // A-matrix (16x4 f32): lanes 0-15 -> K=0,1 = -2x,-2y ;
    //                      lanes 16-31 -> K=2,3 = -2z, 1.0
    const int axoff = hi ? 2 * NTGT : 0;
    float ax = tp[axoff + t0 + m];
    float ay = tp[NTGT + t0 + m];
    v2f a;
    a.x = -2.0f * ax;
    a.y = hi ? 1.0f : (-2.0f * ay);

    float best[8];
    int   bidx[8];
#pragma unroll
    for (int r = 0; r < 8; ++r) { best[r] = 3.4e38f; bidx[r] = 0; }

    for (int c = 0; c < NSRC; c += 16) {
        // One 16B load per lane: {x, y, z, |s|^2} of source s = c + m.
        const v4f p = pk[c];

        // B-matrix (4x16 f32): lanes 0-15 -> K=0,1 = x,y ;
        //                      lanes 16-31 -> K=2,3 = z,|s|^2
        v2f bm;
        bm.x = hi ? p.z : p.x;
        bm.y = hi ? p.w : p.y;

        // D = A*B + 0  ->  |s|^2 - 2 t.s  per (row M, col N); C = inline 0.
        v8f cc = {};
        v8f d = __builtin_amdgcn_wmma_f32_16x16x4_f32(
            /*neg_a=*/false, a, /*neg_b=*/false, bm,
            /*c_mod=*/(short)0, cc, /*reuse_a=*/false, /*reuse_b=*/false);

        const int s = c + m;
#pragma unroll
        for (int r = 0; r < 8; ++r) {
            const bool take = d[r] < best[r];
            best[r] = take ? d[r] : best[r];
            bidx[r] = take ? s    : bidx[r];
        }
    }

    // Butterfly min+argmin across the 16 lanes of each half-wave.
    // Rows 0..7 live in lanes 0..15, rows 8..15 in lanes 16..31.
#pragma unroll
    for (int r = 0; r < 8; ++r) {
        float v  = best[r];
        int   id = bidx[r];
#pragma unroll
        for (int off = 1; off < 16; off <<= 1) {
            float ov = __shfl_xor(v, off, 32);
            int   oi = __shfl_xor(id, off, 32);
            const bool take = (ov < v) || (ov == v && oi < id);
            v  = take ? ov : v;
            id = take ? oi : id;
        }
        best[r] = v;
        bidx[r] = id;
    }

    // Lane 0 writes rows 0..7, lane 16 writes rows 8..15.
    if (m == 0) {
        const int rowBase = hi ? 8 : 0;
#pragma unroll
        for (int r = 0; r < 8; ++r) {
            const float f = ft[bidx[r]];
            const float g = 1.0f / (1.0f + __expf(-f));
            out[(size_t)b * NTGT + t0 + rowBase + r] = g;
        }
    }
}

// ---------------------------------------------------------------------------
extern "C" void kernel_launch(void* const* d_in, const int* in_sizes, int n_in,
                              void* d_out, int out_size, void* d_ws, size_t ws_size,
                              hipStream_t stream) {
    const float* logits = (const float*)d_in[0];   // [B, K, NS]
    const float* spos   = (const float*)d_in[1];   // [B, 3, NS]
    const float* tpos   = (const float*)d_in[2];   // [B, 3, NT]
    float*       out    = (float*)d_out;           // [B, NT]

    // Workspace: packed records (256 KB, 16B-aligned) then feat (64 KB).
    v4f*   packed = (v4f*)d_ws;                          // B*NS * 16B
    float* feat   = (float*)(packed + (size_t)BB * NSRC);

    {
        const int n = BB * NSRC;
        sgg_precompute<<<(n + 255) / 256, 256, 0, stream>>>(logits, spos, feat, packed);
    }
    {
        const int waves   = BB * (NTGT / 16);      // 4096 tiles, 1 wave each
        const int threads = waves * 32;            // 131072
        sgg_nn_gate<<<threads / 256, 256, 0, stream>>>(tpos, packed, feat, out);
    }
}